// SpiralConvWO_77429670412669
// MI455X (gfx1250) — compile-verified
//
#include <hip/hip_runtime.h>

// ---------------------------------------------------------------------------
// Problem constants (from setup_inputs): B=8, C=64, N=4096, K=20, OUT_C=128
// ---------------------------------------------------------------------------
#define BATCH 8
#define CDIM 64
#define NPTS 4096
#define KNN 20
#define OUTC 128
#define FDIM (KNN * CDIM) // 1280

typedef __attribute__((ext_vector_type(2))) float v2f;
typedef __attribute__((ext_vector_type(4))) float v4f;
typedef __attribute__((ext_vector_type(8))) float v8f;

// ---------------------------------------------------------------------------
// Kernel 1: per-point squared norms + transpose x[b,c,n] -> xT[b,n,c]
// ---------------------------------------------------------------------------
__global__ __launch_bounds__(256) void prep_kernel(const float* __restrict__ x,
                                                   float* __restrict__ xx,
                                                   float* __restrict__ xT) {
  const int t = blockIdx.x * 256 + threadIdx.x; // 0 .. B*N-1
  const int b = t >> 12;
  const int n = t & (NPTS - 1);
  const float* xb = x + (size_t)b * CDIM * NPTS + n;
  float s = 0.0f;
#pragma unroll
  for (int c = 0; c < CDIM; ++c) {
    float v = xb[(size_t)c * NPTS]; // coalesced across lanes
    s += v * v;
    xT[(size_t)t * CDIM + c] = v;
  }
  xx[t] = s;
}

// ---------------------------------------------------------------------------
// Kernel 2: pairwise distances via V_WMMA_F32_16X16X4_F32 + fused top-20.
// One wave owns a 16-row i-tile; 8 waves per block; 256 blocks total.
// LDS chunk per wave: 16 rows x 64 cols, padded stride 66 (bank-conflict-free).
// ---------------------------------------------------------------------------
#define CHCOLS 64
#define CHST 66
#define WSLOT 1312 // >= max(16*CHST=1056, 32*KNN*2=1280) floats per wave

__global__ __launch_bounds__(256) void knn_topk_kernel(const float* __restrict__ x,
                                                       const float* __restrict__ xx,
                                                       int* __restrict__ idx_out) {
  __shared__ float smem[8 * WSLOT];

  const int lane = threadIdx.x & 31;
  const int wave = threadIdx.x >> 5;
  const int blk = blockIdx.x;
  const int b = blk >> 5;                       // 32 blocks per batch
  const int i0 = ((blk & 31) * 8 + wave) * 16;  // row-tile base within batch

  const float* xb = x + (size_t)b * CDIM * NPTS;
  const float* xxb = xx + (size_t)b * NPTS;

  const int h = lane >> 4; // 0/1 : selects K pair (A/B frag) and M-half (C/D)
  const int m = lane & 15;

  // Preload A fragments for the whole K=64 panel: af[kk] covers K = kk*4+2h..+1
  float a0[16], a1[16];
#pragma unroll
  for (int kk = 0; kk < 16; ++kk) {
    const int c = kk * 4 + 2 * h;
    a0[kk] = xb[(size_t)c * NPTS + i0 + m];
    a1[kk] = xb[(size_t)(c + 1) * NPTS + i0 + m];
  }
  float xi[8];
#pragma unroll
  for (int r = 0; r < 8; ++r) xi[r] = xxb[i0 + r + 8 * h];

  float* chbuf = smem + wave * WSLOT;

  // Register-resident sorted (descending) top-20 per lane.
  float tv[KNN];
  int ti[KNN];
#pragma unroll
  for (int t = 0; t < KNN; ++t) { tv[t] = -3.0e38f; ti[t] = 0; }

  const int srow = lane >> 1; // scan row (2 lanes per row)
  const int spar = lane & 1;  // column parity

  for (int cb = 0; cb < NPTS; cb += CHCOLS) {
    // ---- compute 4 j-tiles of the chunk with WMMA ----
    for (int jt = 0; jt < CHCOLS; jt += 16) {
      const int j0 = cb + jt;
      v8f acc;
#pragma unroll
      for (int r = 0; r < 8; ++r) acc[r] = 0.0f;
#pragma unroll
      for (int kk = 0; kk < 16; ++kk) {
        const int c = kk * 4 + 2 * h;
        v2f bf;
        bf.x = xb[(size_t)c * NPTS + j0 + m];
        bf.y = xb[(size_t)(c + 1) * NPTS + j0 + m];
        v2f af;
        af.x = a0[kk];
        af.y = a1[kk];
        acc = __builtin_amdgcn_wmma_f32_16x16x4_f32(
            false, af, false, bf, (short)0, acc, false, false);
      }
      const float xj = xxb[j0 + m];
#pragma unroll
      for (int r = 0; r < 8; ++r) {
        // neg squared distance: 2<xi,xj> - |xi|^2 - |xj|^2
        chbuf[(size_t)(r + 8 * h) * CHST + jt + m] = 2.0f * acc[r] - xi[r] - xj;
      }
    }
    // ---- scan chunk: lane handles (row=srow, cols with parity spar) ----
    for (int t = 0; t < CHCOLS / 2; ++t) {
      const int col = 2 * t + spar;
      const float v = chbuf[(size_t)srow * CHST + col];
      if (v > tv[KNN - 1]) {
        tv[KNN - 1] = v;
        ti[KNN - 1] = cb + col;
#pragma unroll
        for (int q = KNN - 1; q > 0; --q) {
          if (tv[q] > tv[q - 1]) {
            const float fv = tv[q]; tv[q] = tv[q - 1]; tv[q - 1] = fv;
            const int fi = ti[q]; ti[q] = ti[q - 1]; ti[q - 1] = fi;
          }
        }
      }
    }
  }

  // ---- merge the two per-row lane lists and emit sorted top-20 indices ----
  __syncthreads();
  float* mv = chbuf;
  int* mi = (int*)(chbuf + 32 * KNN);
#pragma unroll
  for (int t = 0; t < KNN; ++t) {
    mv[lane * KNN + t] = tv[t];
    mi[lane * KNN + t] = ti[t];
  }
  __syncthreads();
  if (spar == 0) {
    const int la = lane * KNN;
    const int lb = (lane + 1) * KNN;
    int pa = 0, pb = 0;
    int* og = idx_out + ((size_t)b * NPTS + i0 + srow) * KNN;
    for (int t = 0; t < KNN; ++t) {
      const float va = mv[la + pa];
      const float vb = mv[lb + pb];
      if (va >= vb) { og[t] = mi[la + pa]; ++pa; }
      else          { og[t] = mi[lb + pb]; ++pb; }
    }
  }
}

// ---------------------------------------------------------------------------
// Kernel 3: gather spirals + output GEMM  out[b,o,n] = bias[o] + A(gather)·Wᵀ
// Block = 16-n tile; 8 waves = 8 o-tiles; K streamed per spiral slot (64).
// ---------------------------------------------------------------------------
#define AST 68 // padded LDS stride for the gathered A chunk

__global__ __launch_bounds__(256) void spiral_gemm_kernel(const float* __restrict__ xT,
                                                          const float* __restrict__ w,
                                                          const float* __restrict__ bias,
                                                          const int* __restrict__ idxb,
                                                          float* __restrict__ out) {
  __shared__ float Ash[16 * AST];

  const int lane = threadIdx.x & 31;
  const int wave = threadIdx.x >> 5;
  const int blk = blockIdx.x;
  const int b = blk >> 8;             // 256 blocks per batch
  const int n0 = (blk & 255) * 16;

  const int h = lane >> 4;
  const int m = lane & 15;
  const int oo = wave * 16 + m; // this lane's output channel

  const int tm = threadIdx.x >> 4; // gather row   (0..15)
  const int tq = threadIdx.x & 15; // gather c/4   (0..15)

  v8f acc;
#pragma unroll
  for (int r = 0; r < 8; ++r) acc[r] = 0.0f;

  for (int s = 0; s < KNN; ++s) {
    __syncthreads();
    const int jm = idxb[((size_t)b * NPTS + n0 + tm) * KNN + s];
    // contiguous 16B gather thanks to xT layout
    const v4f av = *(const v4f*)&xT[((size_t)b * NPTS + jm) * CDIM + tq * 4];
    Ash[tm * AST + tq * 4 + 0] = av.x;
    Ash[tm * AST + tq * 4 + 1] = av.y;
    Ash[tm * AST + tq * 4 + 2] = av.z;
    Ash[tm * AST + tq * 4 + 3] = av.w;
    __syncthreads();
#pragma unroll
    for (int kk = 0; kk < 16; ++kk) {
      const int k0 = kk * 4 + 2 * h;
      const v2f af = *(const v2f*)&Ash[m * AST + k0];
      const v2f bf = *(const v2f*)&w[(size_t)oo * FDIM + s * CDIM + k0];
      acc = __builtin_amdgcn_wmma_f32_16x16x4_f32(
          false, af, false, bf, (short)0, acc, false, false);
    }
  }

  const float bo = bias[oo];
  // lane owns 8 consecutive n for fixed o -> contiguous 32B store
  float* ob = out + ((size_t)b * OUTC + oo) * NPTS + n0 + 8 * h;
#pragma unroll
  for (int r = 0; r < 8; ++r) ob[r] = acc[r] + bo;
}

// ---------------------------------------------------------------------------
// Launch
// ---------------------------------------------------------------------------
extern "C" void kernel_launch(void* const* d_in, const int* in_sizes, int n_in,
                              void* d_out, int out_size, void* d_ws, size_t ws_size,
                              hipStream_t stream) {
  const float* x = (const float*)d_in[0];    // [B, C, N]
  const float* weight = (const float*)d_in[1]; // [OUT_C, K*C]
  const float* bias = (const float*)d_in[2]; // [OUT_C]
  float* out = (float*)d_out;                // [B, OUT_C, N]

  // workspace layout
  float* xx = (float*)d_ws;                                       // B*N floats
  int* idx = (int*)((char*)d_ws + (size_t)BATCH * NPTS * 4);      // B*N*KNN ints
  float* xT = (float*)((char*)d_ws + (size_t)BATCH * NPTS * 4 +
                       (size_t)BATCH * NPTS * KNN * 4);           // B*N*C floats

  prep_kernel<<<(BATCH * NPTS) / 256, 256, 0, stream>>>(x, xx, xT);
  knn_topk_kernel<<<(BATCH * NPTS) / 128, 256, 0, stream>>>(x, xx, idx);
  spiral_gemm_kernel<<<BATCH * (NPTS / 16), 256, 0, stream>>>(xT, weight, bias, idx, out);
}